// ProposalModule_84705345012399
// MI455X (gfx1250) — compile-verified
//
#include <hip/hip_runtime.h>
#include <math.h>

// ---------------- problem constants ----------------
#define BATCH   16
#define KPTS    4096
#define CSEED   256
#define NPOINT  512
#define NSAMPLE 16
#define RADIUS  0.3f
#define OUTCH   101                     // 2+3+6+18*4+18
#define MROWS   (BATCH*NPOINT*NSAMPLE)  // 131072
#define MPTS    (BATCH*NPOINT)          // 8192
#define EPS_BN  1e-5f
#define OUT_F   212

typedef __attribute__((ext_vector_type(2))) float v2f;
typedef __attribute__((ext_vector_type(8))) float v8f;

// =====================================================================
// 1) Farthest point sampling: one workgroup per batch, 512 threads.
//    dists kept in LDS; block argmax with first-index tie-break
//    (matches jnp.argmax semantics).
// =====================================================================
__global__ void fps_kernel(const float* __restrict__ xyz,
                           float* __restrict__ new_xyz,
                           int* __restrict__ fps_inds) {
    __shared__ float dist[KPTS];
    __shared__ float sval[512];
    __shared__ int   sidx[512];
    const int b = blockIdx.x, t = threadIdx.x;
    const float* xb = xyz + (size_t)b * KPTS * 3;

    for (int k = t; k < KPTS; k += 512) dist[k] = 1e10f;
    __syncthreads();

    int last = 0;
    if (t == 0) {
        fps_inds[b * NPOINT] = 0;
        float* nz = new_xyz + (size_t)b * NPOINT * 3;
        nz[0] = xb[0]; nz[1] = xb[1]; nz[2] = xb[2];
    }
    for (int it = 1; it < NPOINT; ++it) {
        const float px = xb[last * 3 + 0];
        const float py = xb[last * 3 + 1];
        const float pz = xb[last * 3 + 2];
        for (int k = t; k < KPTS; k += 512) {
            float dx = xb[k * 3 + 0] - px;
            float dy = xb[k * 3 + 1] - py;
            float dz = xb[k * 3 + 2] - pz;
            float d  = dx * dx + dy * dy + dz * dz;
            if (d < dist[k]) dist[k] = d;
        }
        __syncthreads();
        float bv = -1.0f; int bi = 0;
        for (int k = t; k < KPTS; k += 512) {
            float v = dist[k];
            if (v > bv) { bv = v; bi = k; }   // ascending scan => first max kept
        }
        sval[t] = bv; sidx[t] = bi;
        __syncthreads();
        for (int s = 256; s > 0; s >>= 1) {
            if (t < s) {
                float ov = sval[t + s]; int oi = sidx[t + s];
                if (ov > sval[t] || (ov == sval[t] && oi < sidx[t])) {
                    sval[t] = ov; sidx[t] = oi;
                }
            }
            __syncthreads();
        }
        last = sidx[0];
        if (t == 0) {
            fps_inds[b * NPOINT + it] = last;
            float* nz = new_xyz + ((size_t)b * NPOINT + it) * 3;
            nz[0] = xb[last * 3 + 0];
            nz[1] = xb[last * 3 + 1];
            nz[2] = xb[last * 3 + 2];
        }
        __syncthreads();
    }
}

// =====================================================================
// 2) Ball query: first NSAMPLE indices (ascending) with dist2 < r^2,
//    padded with the first hit. xyz cached in LDS (48 KB).
// =====================================================================
__global__ void ball_kernel(const float* __restrict__ xyz,
                            const float* __restrict__ new_xyz,
                            int* __restrict__ idxbuf) {
    __shared__ float xs[KPTS], ys[KPTS], zs[KPTS];
    const int b = blockIdx.x >> 3;         // 8 blocks per batch
    const int chunk = blockIdx.x & 7;
    const int t = threadIdx.x;             // 64 threads
    const float* xb = xyz + (size_t)b * KPTS * 3;
    for (int k = t; k < KPTS; k += 64) {
        xs[k] = xb[k * 3 + 0];
        ys[k] = xb[k * 3 + 1];
        zs[k] = xb[k * 3 + 2];
    }
    __syncthreads();

    const int q = chunk * 64 + t;          // 0..511
    const float* nz = new_xyz + ((size_t)b * NPOINT + q) * 3;
    const float px = nz[0], py = nz[1], pz = nz[2];
    int* out = idxbuf + ((size_t)b * NPOINT + q) * NSAMPLE;
    const float r2 = RADIUS * RADIUS;
    int cnt = 0, first = 0;
    for (int k = 0; k < KPTS && cnt < NSAMPLE; ++k) {
        float dx = px - xs[k], dy = py - ys[k], dz = pz - zs[k];
        float d = dx * dx + dy * dy + dz * dz;
        if (d < r2) {
            if (cnt == 0) first = k;
            out[cnt++] = k;
        }
    }
    for (int j = cnt; j < NSAMPLE; ++j) out[j] = first;
}

// =====================================================================
// 3) Weight transpose + zero pad: src is (out x in) row-major,
//    dst is (kpad x npad) with dst[k][n] = src[n][k].
// =====================================================================
__global__ void wtrans_kernel(const float* __restrict__ src, float* __restrict__ dst,
                              int kin, int nout, int kpad, int npad) {
    int i = blockIdx.x * blockDim.x + threadIdx.x;
    int tot = kpad * npad;
    if (i >= tot) return;
    int k = i / npad, n = i % npad;
    dst[i] = (k < kin && n < nout) ? src[n * kin + k] : 0.0f;
}

// =====================================================================
// 4) Layer-0 GEMM with fused neighbor gather.
//    A row m = (bp*16 + s): cols = [g_xyz(3) | features(256) | pad] (K=260)
//    Each wave: one query point, 16 rows x 128 cols via 8 accumulators.
//    Output Yt is channel-major (128 x MROWS).
// =====================================================================
__global__ void gemm0_kernel(const float* __restrict__ xyz,
                             const float* __restrict__ feats,
                             const float* __restrict__ new_xyz,
                             const int* __restrict__ idxbuf,
                             const float* __restrict__ Bt,   // 260 x 128
                             float* __restrict__ Yt) {       // 128 x MROWS
    const int lane = threadIdx.x & 31;
    const int wave = threadIdx.x >> 5;
    const int m0 = (blockIdx.x * 4 + wave) * 16;
    const int bp = m0 >> 4;                  // one (b,p) per wave
    const int b  = bp / NPOINT;
    const int s  = lane & 15;
    const int kk = (lane >> 4) ? 2 : 0;
    const int col = lane & 15;

    const int myidx = idxbuf[(size_t)bp * NSAMPLE + s];
    const float* nz = new_xyz + (size_t)bp * 3;
    const float* pc = xyz + ((size_t)b * KPTS + myidx) * 3;
    const float inv_r = 1.0f / RADIUS;
    const float gx0 = (pc[0] - nz[0]) * inv_r;
    const float gx1 = (pc[1] - nz[1]) * inv_r;
    const float gx2 = (pc[2] - nz[2]) * inv_r;
    const float* fb = feats + (size_t)b * CSEED * KPTS + myidx;  // feat(c)=fb[c*KPTS]

    v8f zero = {};
    v8f acc[8];
#pragma unroll
    for (int t = 0; t < 8; ++t) acc[t] = zero;

    for (int k = 0; k < 260; k += 4) {
        v2f a;
        if (k == 0) {
            a.x = (kk == 0) ? gx0 : gx2;
            a.y = (kk == 0) ? gx1 : fb[0];
        } else {
            int c = k + kk - 3;                       // 1..255
            a.x = fb[(size_t)c * KPTS];
            a.y = (c + 1 < CSEED) ? fb[(size_t)(c + 1) * KPTS] : 0.0f;
        }
        const float* brow = Bt + (size_t)(k + kk) * 128;
#pragma unroll
        for (int t = 0; t < 8; ++t) {
            v2f bb;
            bb.x = brow[t * 16 + col];
            bb.y = brow[128 + t * 16 + col];
            acc[t] = __builtin_amdgcn_wmma_f32_16x16x4_f32(
                false, a, false, bb, (short)0, acc[t], false, false);
        }
    }
    const int mlo = m0 + ((lane >> 4) ? 8 : 0);
#pragma unroll
    for (int r = 0; r < 8; ++r) {
        int m = mlo + r;
#pragma unroll
        for (int t = 0; t < 8; ++t)
            Yt[(size_t)(t * 16 + col) * MROWS + m] = acc[t][r];
    }
}

// =====================================================================
// 5) Generic GEMM: At (Kpad x M, channel-major, coalesced A loads),
//    Bt (Kpad x NT*16, coalesced B loads), Yt (N x M channel-major).
// =====================================================================
template <int NT>
__global__ void gemm_kernel(const float* __restrict__ At,
                            const float* __restrict__ Bt,
                            float* __restrict__ Yt,
                            int M, int N, int Kpad) {
    const int lane = threadIdx.x & 31;
    const int wave = threadIdx.x >> 5;
    const int m0 = (blockIdx.x * 4 + wave) * 16;
    const int kk = (lane >> 4) ? 2 : 0;
    const int col = lane & 15;
    const int row = m0 + col;
    const int ldb = NT * 16;

    v8f zero = {};
    v8f acc[NT];
#pragma unroll
    for (int t = 0; t < NT; ++t) acc[t] = zero;

    for (int k = 0; k < Kpad; k += 4) {
        v2f a;
        a.x = At[(size_t)(k + kk) * M + row];
        a.y = At[(size_t)(k + kk + 1) * M + row];
        const float* brow = Bt + (size_t)(k + kk) * ldb;
#pragma unroll
        for (int t = 0; t < NT; ++t) {
            v2f bb;
            bb.x = brow[t * 16 + col];
            bb.y = brow[ldb + t * 16 + col];
            acc[t] = __builtin_amdgcn_wmma_f32_16x16x4_f32(
                false, a, false, bb, (short)0, acc[t], false, false);
        }
    }
    const int mlo = m0 + ((lane >> 4) ? 8 : 0);
#pragma unroll
    for (int r = 0; r < 8; ++r) {
        int m = mlo + r;
#pragma unroll
        for (int t = 0; t < NT; ++t) {
            int n = t * 16 + col;
            if (n < N) Yt[(size_t)n * M + m] = acc[t][r];
        }
    }
}

// =====================================================================
// 6) BN statistics: one block per channel, contiguous row reduction.
// =====================================================================
__global__ void bn_stats_kernel(const float* __restrict__ Yt,
                                float* __restrict__ stats, int M) {
    __shared__ float s1[256], s2[256];
    const int c = blockIdx.x, t = threadIdx.x;
    const float* p = Yt + (size_t)c * M;
    float a = 0.0f, b = 0.0f;
    for (int i = t; i < M; i += 256) {
        float v = p[i];
        a += v; b += v * v;
    }
    s1[t] = a; s2[t] = b;
    __syncthreads();
    for (int s = 128; s > 0; s >>= 1) {
        if (t < s) { s1[t] += s1[t + s]; s2[t] += s2[t + s]; }
        __syncthreads();
    }
    if (t == 0) { stats[2 * c] = s1[0]; stats[2 * c + 1] = s2[0]; }
}

// =====================================================================
// 7) BN normalize + ReLU (in place). M is a power of two: mshift=log2(M).
// =====================================================================
__global__ void bn_relu_kernel(float* __restrict__ Yt,
                               const float* __restrict__ stats,
                               const float* __restrict__ gamma,
                               const float* __restrict__ beta,
                               int mshift, float invM, size_t tot) {
    size_t i = (size_t)blockIdx.x * 256 + threadIdx.x;
    if (i >= tot) return;
    int c = (int)(i >> mshift);
    float mu  = stats[2 * c] * invM;
    float var = stats[2 * c + 1] * invM - mu * mu;
    float sc  = gamma[c] * rsqrtf(var + EPS_BN);
    float v   = (Yt[i] - mu) * sc + beta[c];
    Yt[i] = v > 0.0f ? v : 0.0f;
}

// =====================================================================
// 8) Max pool over the 16 samples of each query point.
// =====================================================================
__global__ void maxpool_kernel(const float* __restrict__ Yt,
                               float* __restrict__ aggT) {
    int i = blockIdx.x * 256 + threadIdx.x;   // c*MPTS + bp
    if (i >= 128 * MPTS) return;
    int c = i / MPTS, bp = i % MPTS;
    const float* p = Yt + (size_t)c * MROWS + (size_t)bp * NSAMPLE;
    float m = p[0];
#pragma unroll
    for (int s = 1; s < NSAMPLE; ++s) m = fmaxf(m, p[s]);
    aggT[i] = m;
}

// =====================================================================
// 9) Head: bias add, rotation basis (Gram-Schmidt via crosses),
//    size residual expansion, output concat (B,512,212).
// =====================================================================
__global__ void head_kernel(const float* __restrict__ yfT,
                            const float* __restrict__ pbias,
                            const float* __restrict__ new_xyz,
                            const float* __restrict__ msize,
                            float* __restrict__ out) {
    int bp = blockIdx.x * blockDim.x + threadIdx.x;
    if (bp >= MPTS) return;
    float net[OUTCH];
#pragma unroll
    for (int c = 0; c < OUTCH; ++c)
        net[c] = yfT[(size_t)c * MPTS + bp] + pbias[c];

    float* o = out + (size_t)bp * OUT_F;
    o[0] = net[0]; o[1] = net[1];
    const float* nz = new_xyz + (size_t)bp * 3;
    o[2] = nz[0] + net[2]; o[3] = nz[1] + net[3]; o[4] = nz[2] + net[4];

    float xr0 = net[5], xr1 = net[6], xr2 = net[7];
    float yr0 = net[8], yr1 = net[9], yr2 = net[10];
    float yn = sqrtf(yr0 * yr0 + yr1 * yr1 + yr2 * yr2) + 1e-8f;
    float y0 = yr0 / yn, y1 = yr1 / yn, y2 = yr2 / yn;
    float z0 = xr1 * y2 - xr2 * y1;
    float z1 = xr2 * y0 - xr0 * y2;
    float z2 = xr0 * y1 - xr1 * y0;
    float zn = sqrtf(z0 * z0 + z1 * z1 + z2 * z2) + 1e-8f;
    z0 /= zn; z1 /= zn; z2 /= zn;
    float x0 = y1 * z2 - y2 * z1;
    float x1 = y2 * z0 - y0 * z2;
    float x2 = y0 * z1 - y1 * z0;
    o[5]  = x0; o[6]  = y0; o[7]  = z0;
    o[8]  = x1; o[9]  = y1; o[10] = z1;
    o[11] = x2; o[12] = y2; o[13] = z2;

#pragma unroll
    for (int j = 0; j < 18; ++j) o[14 + j] = net[11 + j];          // ss
#pragma unroll
    for (int j = 0; j < 54; ++j) {                                  // srn/sr/sc
        float v  = net[29 + j];
        float ms = msize[j];
        o[32 + j]  = v;
        o[86 + j]  = v * ms;
        o[140 + j] = (v + 1.0f) * ms;
    }
#pragma unroll
    for (int j = 0; j < 18; ++j) o[194 + j] = net[83 + j];         // sem
}

// =====================================================================
// launcher
// =====================================================================
extern "C" void kernel_launch(void* const* d_in, const int* in_sizes, int n_in,
                              void* d_out, int out_size, void* d_ws, size_t ws_size,
                              hipStream_t stream) {
    (void)in_sizes; (void)n_in; (void)out_size; (void)ws_size;
    const float* xyz    = (const float*)d_in[0];
    const float* feats  = (const float*)d_in[1];
    const float* w0     = (const float*)d_in[2];
    const float* g0     = (const float*)d_in[3];
    const float* b0     = (const float*)d_in[4];
    const float* w1     = (const float*)d_in[5];
    const float* g1     = (const float*)d_in[6];
    const float* b1     = (const float*)d_in[7];
    const float* w2     = (const float*)d_in[8];
    const float* g2     = (const float*)d_in[9];
    const float* b2     = (const float*)d_in[10];
    const float* pw0    = (const float*)d_in[11];
    const float* pg0    = (const float*)d_in[12];
    const float* pb0    = (const float*)d_in[13];
    const float* pw1    = (const float*)d_in[14];
    const float* pg1    = (const float*)d_in[15];
    const float* pb1    = (const float*)d_in[16];
    const float* pw2    = (const float*)d_in[17];
    const float* pbias  = (const float*)d_in[18];
    const float* msize  = (const float*)d_in[19];
    float* out = (float*)d_out;

    char* base = (char*)d_ws;
    size_t off = 0;
    auto take = [&](size_t bytes) -> char* {
        char* p = base + off;
        off += (bytes + 255) & ~(size_t)255;
        return p;
    };
    float* newxyz = (float*)take((size_t)BATCH * NPOINT * 3 * 4);
    int*   fpsi   = (int*)  take((size_t)BATCH * NPOINT * 4);
    int*   idxbuf = (int*)  take((size_t)BATCH * NPOINT * NSAMPLE * 4);
    float* W0t    = (float*)take(260 * 128 * 4);
    float* W1t    = (float*)take(128 * 128 * 4);
    float* W2t    = (float*)take(128 * 128 * 4);
    float* P0t    = (float*)take(128 * 128 * 4);
    float* P1t    = (float*)take(128 * 128 * 4);
    float* P2t    = (float*)take(128 * 112 * 4);
    float* buf1   = (float*)take((size_t)128 * MROWS * 4);
    float* buf2   = (float*)take((size_t)128 * MROWS * 4);
    float* aggT   = (float*)take((size_t)128 * MPTS * 4);
    float* pbuf1  = (float*)take((size_t)128 * MPTS * 4);
    float* pbuf2  = (float*)take((size_t)128 * MPTS * 4);
    float* yfT    = (float*)take((size_t)112 * MPTS * 4);
    float* stats  = (float*)take(256 * 4);

    // ---- sampling / grouping ----
    fps_kernel <<<BATCH, 512, 0, stream>>>(xyz, newxyz, fpsi);
    ball_kernel<<<BATCH * 8, 64, 0, stream>>>(xyz, newxyz, idxbuf);

    // ---- weight transposes (K-major, zero padded) ----
    wtrans_kernel<<<(260 * 128 + 255) / 256, 256, 0, stream>>>(w0,  W0t, 259, 128, 260, 128);
    wtrans_kernel<<<(128 * 128 + 255) / 256, 256, 0, stream>>>(w1,  W1t, 128, 128, 128, 128);
    wtrans_kernel<<<(128 * 128 + 255) / 256, 256, 0, stream>>>(w2,  W2t, 128, 128, 128, 128);
    wtrans_kernel<<<(128 * 128 + 255) / 256, 256, 0, stream>>>(pw0, P0t, 128, 128, 128, 128);
    wtrans_kernel<<<(128 * 128 + 255) / 256, 256, 0, stream>>>(pw1, P1t, 128, 128, 128, 128);
    wtrans_kernel<<<(128 * 112 + 255) / 256, 256, 0, stream>>>(pw2, P2t, 128, OUTCH, 128, 112);

    const size_t tot0 = (size_t)128 * MROWS;
    const size_t tot1 = (size_t)128 * MPTS;

    // ---- grouped MLP (3 layers, BN over B*NP*NS) ----
    gemm0_kernel<<<MROWS / 64, 128, 0, stream>>>(xyz, feats, newxyz, idxbuf, W0t, buf1);
    bn_stats_kernel<<<128, 256, 0, stream>>>(buf1, stats, MROWS);
    bn_relu_kernel<<<(unsigned)(tot0 / 256), 256, 0, stream>>>(buf1, stats, g0, b0, 17, 1.0f / MROWS, tot0);

    gemm_kernel<8><<<MROWS / 64, 128, 0, stream>>>(buf1, W1t, buf2, MROWS, 128, 128);
    bn_stats_kernel<<<128, 256, 0, stream>>>(buf2, stats, MROWS);
    bn_relu_kernel<<<(unsigned)(tot0 / 256), 256, 0, stream>>>(buf2, stats, g1, b1, 17, 1.0f / MROWS, tot0);

    gemm_kernel<8><<<MROWS / 64, 128, 0, stream>>>(buf2, W2t, buf1, MROWS, 128, 128);
    bn_stats_kernel<<<128, 256, 0, stream>>>(buf1, stats, MROWS);
    bn_relu_kernel<<<(unsigned)(tot0 / 256), 256, 0, stream>>>(buf1, stats, g2, b2, 17, 1.0f / MROWS, tot0);

    // ---- max pool over samples ----
    maxpool_kernel<<<(128 * MPTS + 255) / 256, 256, 0, stream>>>(buf1, aggT);

    // ---- proposal MLP (2 layers, BN over B*NP) ----
    gemm_kernel<8><<<MPTS / 64, 128, 0, stream>>>(aggT, P0t, pbuf1, MPTS, 128, 128);
    bn_stats_kernel<<<128, 256, 0, stream>>>(pbuf1, stats, MPTS);
    bn_relu_kernel<<<(unsigned)(tot1 / 256), 256, 0, stream>>>(pbuf1, stats, pg0, pb0, 13, 1.0f / MPTS, tot1);

    gemm_kernel<8><<<MPTS / 64, 128, 0, stream>>>(pbuf1, P1t, pbuf2, MPTS, 128, 128);
    bn_stats_kernel<<<128, 256, 0, stream>>>(pbuf2, stats, MPTS);
    bn_relu_kernel<<<(unsigned)(tot1 / 256), 256, 0, stream>>>(pbuf2, stats, pg1, pb1, 13, 1.0f / MPTS, tot1);

    // ---- final projection (101 outputs -> 7 N-tiles) + head ----
    gemm_kernel<7><<<MPTS / 64, 128, 0, stream>>>(pbuf2, P2t, yfT, MPTS, OUTCH, 128);
    head_kernel<<<MPTS / 256, 256, 0, stream>>>(yfT, pbias, newxyz, msize, out);
}